// MultiHeadAttentionBase_65481071395643
// MI455X (gfx1250) — compile-verified
//
#include <hip/hip_runtime.h>

typedef __attribute__((ext_vector_type(2))) float v2f;
typedef __attribute__((ext_vector_type(8))) float v8f;

namespace {
constexpr int S_LEN = 2048;
constexpr int DK    = 4;
constexpr int NKT   = S_LEN / 16;      // 128 key tiles
constexpr int BH    = 64;              // B*H
constexpr float NEG_BIG = -1.0e9f;
}

// ds_swizzle XOR butterfly within 32 lanes: new_lane = lane ^ XOR (and=0x1F, or=0)
template <int OFF>
__device__ __forceinline__ float dsswz(float x) {
    return __int_as_float(__builtin_amdgcn_ds_swizzle(__float_as_int(x), OFF));
}
__device__ __forceinline__ float red_max16(float x) {  // xor 1,2,4,8: stays within 16-lane half
    x = fmaxf(x, dsswz<0x041f>(x));
    x = fmaxf(x, dsswz<0x081f>(x));
    x = fmaxf(x, dsswz<0x101f>(x));
    x = fmaxf(x, dsswz<0x201f>(x));
    return x;
}
__device__ __forceinline__ float red_sum16(float x) {
    x += dsswz<0x041f>(x);
    x += dsswz<0x081f>(x);
    x += dsswz<0x101f>(x);
    x += dsswz<0x201f>(x);
    return x;
}

__global__ __launch_bounds__(256) void mha_fused_wmma(
    const float* __restrict__ Q, const float* __restrict__ Kp,
    const float* __restrict__ Vp, const int* __restrict__ Mp,
    float* __restrict__ Out, float* __restrict__ Attn)
{
    const int lane = threadIdx.x & 31;
    const int wid  = (blockIdx.x * blockDim.x + threadIdx.x) >> 5;
    const int bh   = wid >> 7;              // 0..63
    const int qt   = wid & (NKT - 1);       // query tile 0..127
    const int b    = bh >> 4;
    const int l16  = lane & 15;
    const int hi   = lane >> 4;             // 0 or 1
    const int q0   = qt * 16;

    const float* qb = Q  + (size_t)bh * S_LEN * DK;
    const float* kb = Kp + (size_t)bh * S_LEN * DK;
    const float* vb = Vp + (size_t)bh * S_LEN * DK;
    const int*   mb = Mp + (size_t)b * S_LEN * S_LEN + (size_t)q0 * S_LEN;

    // A fragment (16x4 fp32), pre-scaled by 1/sqrt(d)=0.5 (exact, power of two):
    // lane l<16 holds Q[q0+l][0..1], lane l>=16 holds Q[q0+l-16][2..3]
    v2f a = *(const v2f*)(qb + (size_t)(q0 + l16) * DK + 2 * hi);
    a.x *= 0.5f;
    a.y *= 0.5f;

    // ---------------- pass 1: per-row max (lane-local, reduce once) --------
    float mrow[8];
#pragma unroll
    for (int r = 0; r < 8; ++r) mrow[r] = -3.0e38f;

    for (int kt = 0; kt < NKT; ++kt) {
        const int k0 = kt * 16;
        const v2f bf = *(const v2f*)(kb + (size_t)(k0 + l16) * DK + 2 * hi);
        v8f c = {};
        c = __builtin_amdgcn_wmma_f32_16x16x4_f32(false, a, false, bf,
                                                  (short)0, c, false, false);
#pragma unroll
        for (int r = 0; r < 8; ++r) {
            const int mi = r + 8 * hi;                       // query row in tile
            const int mk = mb[(size_t)mi * S_LEN + (k0 + l16)];
            const float sv = mk ? c[r] : NEG_BIG;
            mrow[r] = fmaxf(mrow[r], sv);
        }
    }
#pragma unroll
    for (int r = 0; r < 8; ++r) mrow[r] = red_max16(mrow[r]);

    // ---------------- pass 2: sum(exp) and unnormalized O ------------------
    float lsum[8];
    float oacc[8][4];
#pragma unroll
    for (int r = 0; r < 8; ++r) {
        lsum[r] = 0.0f;
#pragma unroll
        for (int j = 0; j < 4; ++j) oacc[r][j] = 0.0f;
    }

    for (int kt = 0; kt < NKT; ++kt) {
        const int k0 = kt * 16;
        const v2f bf = *(const v2f*)(kb + (size_t)(k0 + l16) * DK + 2 * hi);
        v8f c = {};
        c = __builtin_amdgcn_wmma_f32_16x16x4_f32(false, a, false, bf,
                                                  (short)0, c, false, false);
        const float4 vv = *(const float4*)(vb + (size_t)(k0 + l16) * DK);
#pragma unroll
        for (int r = 0; r < 8; ++r) {
            const int mi = r + 8 * hi;
            const int mk = mb[(size_t)mi * S_LEN + (k0 + l16)];
            const float sv = mk ? c[r] : NEG_BIG;
            const float e  = __expf(sv - mrow[r]);
            lsum[r]    += e;
            oacc[r][0] += e * vv.x;
            oacc[r][1] += e * vv.y;
            oacc[r][2] += e * vv.z;
            oacc[r][3] += e * vv.w;
        }
    }
#pragma unroll
    for (int r = 0; r < 8; ++r) {
        lsum[r]    = red_sum16(lsum[r]);
        oacc[r][0] = red_sum16(oacc[r][0]);
        oacc[r][1] = red_sum16(oacc[r][1]);
        oacc[r][2] = red_sum16(oacc[r][2]);
        oacc[r][3] = red_sum16(oacc[r][3]);
    }
    float rinv[8];
#pragma unroll
    for (int r = 0; r < 8; ++r) rinv[r] = 1.0f / lsum[r];

    // ---------------- pass 3: stream out normalized attention --------------
    float* attn_q = Attn + ((size_t)bh * S_LEN + q0) * S_LEN;
    for (int kt = 0; kt < NKT; ++kt) {
        const int k0 = kt * 16;
        const v2f bf = *(const v2f*)(kb + (size_t)(k0 + l16) * DK + 2 * hi);
        v8f c = {};
        c = __builtin_amdgcn_wmma_f32_16x16x4_f32(false, a, false, bf,
                                                  (short)0, c, false, false);
#pragma unroll
        for (int r = 0; r < 8; ++r) {
            const int mi = r + 8 * hi;
            const int mk = mb[(size_t)mi * S_LEN + (k0 + l16)];
            const float sv = mk ? c[r] : NEG_BIG;
            const float p  = __expf(sv - mrow[r]) * rinv[r];
            __builtin_nontemporal_store(p, attn_q + (size_t)mi * S_LEN + (k0 + l16));
        }
    }

    // ---------------- write O (divergent, after all WMMAs) -----------------
#pragma unroll
    for (int r = 0; r < 8; ++r) {
        const int mi = r + 8 * hi;
        const float ov = (l16 == 0) ? oacc[r][0]
                       : (l16 == 1) ? oacc[r][1]
                       : (l16 == 2) ? oacc[r][2]
                       :              oacc[r][3];
        if (l16 < 4)
            Out[((size_t)bh * S_LEN + q0 + mi) * DK + l16] = ov * rinv[r];
    }
}

extern "C" void kernel_launch(void* const* d_in, const int* in_sizes, int n_in,
                              void* d_out, int out_size, void* d_ws, size_t ws_size,
                              hipStream_t stream) {
    (void)in_sizes; (void)n_in; (void)out_size; (void)d_ws; (void)ws_size;
    const float* Q = (const float*)d_in[0];
    const float* K = (const float*)d_in[1];
    const float* V = (const float*)d_in[2];
    const int*   M = (const int*)d_in[3];

    float* Out  = (float*)d_out;
    float* Attn = Out + (size_t)BH * S_LEN * DK;   // attn follows out in tuple order

    const int total_waves     = BH * NKT;          // 8192
    const int waves_per_block = 8;                 // 256 threads
    dim3 grid(total_waves / waves_per_block), block(256);
    mha_fused_wmma<<<grid, block, 0, stream>>>(Q, K, V, M, Out, Attn);
}